// LuongeAttention_1881195675868
// MI455X (gfx1250) — compile-verified
//
#include <hip/hip_runtime.h>
#include <hip/hip_bf16.h>

#define B 32
#define T 2048
#define H 1024
#define U 1024
#define NEG_BIG -1e9f

typedef float v2f __attribute__((ext_vector_type(2)));
typedef float v8f __attribute__((ext_vector_type(8)));

// ---------------------------------------------------------------------------
// Kernel 1: v[b][h] = sum_u dec[b][u] * Wa[h][u]   (dec: B x U, Wa: H x U)
// WMMA f32 16x16x4. Block = 256 threads (8 waves), each wave owns one
// 16(b) x 16(h) D-tile; block covers all 32 b x 64 h. grid.x = H/64 = 16.
// LDS staging with row stride 36 floats (144B) -> conflict-free frag reads.
// ---------------------------------------------------------------------------
__global__ __launch_bounds__(256) void k_project(const float* __restrict__ dec,
                                                 const float* __restrict__ Wa,
                                                 float* __restrict__ vout) {
  __shared__ float lds_dec[32 * 36];  // [b][u-chunk]
  __shared__ float lds_wa[64 * 36];   // [h][u-chunk]
  const int t    = threadIdx.x;
  const int wave = t >> 5;
  const int lane = t & 31;
  const int hblk = blockIdx.x * 64;
  const int m0   = (wave & 1) * 16;   // b tile
  const int n0   = (wave >> 1) * 16;  // h tile (within block's 64)
  const int lrow = lane & 15;
  const int khalf = (lane >> 4) * 2;  // lanes 0-15: K={0,1}; lanes 16-31: K={2,3}

  v8f acc = {};
  for (int u0 = 0; u0 < U; u0 += 32) {
    // stage dec chunk: 32 rows x 32 u (coalesced float4)
    {
      const int r = t >> 3, c4 = (t & 7) * 4;
      const float4 d = *(const float4*)(dec + r * U + u0 + c4);
      *(float4*)(&lds_dec[r * 36 + c4]) = d;
    }
    // stage Wa chunk: 64 rows x 32 u (2 float4 per thread, coalesced)
#pragma unroll
    for (int i = 0; i < 2; ++i) {
      const int idx = t + i * 256;
      const int r = idx >> 3, c4 = (idx & 7) * 4;
      const float4 d = *(const float4*)(Wa + (size_t)(hblk + r) * U + u0 + c4);
      *(float4*)(&lds_wa[r * 36 + c4]) = d;
    }
    __syncthreads();
#pragma unroll
    for (int ku = 0; ku < 32; ku += 4) {
      const v2f a = *(const v2f*)(&lds_dec[(m0 + lrow) * 36 + ku + khalf]);
      const v2f bf = *(const v2f*)(&lds_wa[(n0 + lrow) * 36 + ku + khalf]);
      acc = __builtin_amdgcn_wmma_f32_16x16x4_f32(false, a, false, bf,
                                                  (short)0, acc, false, false);
    }
    __syncthreads();
  }
  // D layout: lane L, VGPR r -> (m = r + 8*(L/16), n = L%16)
  const int bb = m0 + (lane >> 4) * 8;
  const int hh = hblk + n0 + lrow;
#pragma unroll
  for (int r = 0; r < 8; ++r) {
    vout[(size_t)(bb + r) * H + hh] = acc[r];
  }
}

// ---------------------------------------------------------------------------
// Kernel 2: score[b][t] = mask ? dot(enc[b,t,:], v[b,:]) : -1e9
// grid = (T/64, B), block 256 (8 waves). One wave per row; v[b] in LDS.
// ---------------------------------------------------------------------------
__global__ __launch_bounds__(256) void k_score(const float* __restrict__ enc,
                                               const float* __restrict__ v,
                                               const unsigned char* __restrict__ mask,
                                               float* __restrict__ score) {
  __shared__ float lds_v[H];
  const int b  = blockIdx.y;
  const int t0 = blockIdx.x * 64;
  for (int i = threadIdx.x; i < H; i += 256) lds_v[i] = v[(size_t)b * H + i];
  __syncthreads();
  const int wave = threadIdx.x >> 5;
  const int lane = threadIdx.x & 31;
#pragma unroll 1
  for (int rt = 0; rt < 8; ++rt) {
    const int tt = t0 + wave * 8 + rt;
    const float* row = enc + ((size_t)b * T + tt) * H;
    float acc = 0.f;
#pragma unroll
    for (int k = 0; k < 8; ++k) {
      const float4 e = *(const float4*)(row + k * 128 + lane * 4);
      const float4 w = *(const float4*)(&lds_v[k * 128 + lane * 4]);
      acc += e.x * w.x + e.y * w.y + e.z * w.z + e.w * w.w;
    }
#pragma unroll
    for (int off = 16; off > 0; off >>= 1) acc += __shfl_xor(acc, off, 32);
    if (lane == 0) {
      score[(size_t)b * T + tt] = mask[(size_t)b * T + tt] ? acc : NEG_BIG;
    }
  }
}

// ---------------------------------------------------------------------------
// Kernel 3: softmax over T per batch row. grid = B, block 256.
// ---------------------------------------------------------------------------
__global__ __launch_bounds__(256) void k_softmax(const float* __restrict__ score,
                                                 float* __restrict__ attn) {
  __shared__ float red[256];
  const int b = blockIdx.x;
  float m = -INFINITY;
  for (int t = threadIdx.x; t < T; t += 256) m = fmaxf(m, score[(size_t)b * T + t]);
  red[threadIdx.x] = m;
  __syncthreads();
  for (int s = 128; s > 0; s >>= 1) {
    if (threadIdx.x < s) red[threadIdx.x] = fmaxf(red[threadIdx.x], red[threadIdx.x + s]);
    __syncthreads();
  }
  m = red[0];
  __syncthreads();
  float sum = 0.f;
  for (int t = threadIdx.x; t < T; t += 256) sum += __expf(score[(size_t)b * T + t] - m);
  red[threadIdx.x] = sum;
  __syncthreads();
  for (int s = 128; s > 0; s >>= 1) {
    if (threadIdx.x < s) red[threadIdx.x] += red[threadIdx.x + s];
    __syncthreads();
  }
  const float inv = 1.f / red[0];
  for (int t = threadIdx.x; t < T; t += 256)
    attn[(size_t)b * T + t] = __expf(score[(size_t)b * T + t] - m) * inv;
}

// ---------------------------------------------------------------------------
// Kernel 4: partial context. grid = (H/256, T/128, B), block 256.
// partial[b][tc][h] = sum_{t in chunk} attn[b,t]*mask[b,t]*enc[b,t,h]
// ---------------------------------------------------------------------------
__global__ __launch_bounds__(256) void k_ctx_partial(const float* __restrict__ enc,
                                                     const float* __restrict__ attn,
                                                     const unsigned char* __restrict__ mask,
                                                     float* __restrict__ partial) {
  __shared__ float w[128];
  const int hc = blockIdx.x, tc = blockIdx.y, b = blockIdx.z;
  const int h  = hc * 256 + threadIdx.x;
  const int t0 = tc * 128;
  if (threadIdx.x < 128) {
    const size_t ti = (size_t)b * T + t0 + threadIdx.x;
    w[threadIdx.x] = attn[ti] * (mask[ti] ? 1.f : 0.f);
  }
  __syncthreads();
  float acc = 0.f;
#pragma unroll 4
  for (int i = 0; i < 128; ++i) {
    acc += w[i] * enc[((size_t)b * T + t0 + i) * H + h];
  }
  partial[((size_t)b * 16 + tc) * H + h] = acc;
}

// ---------------------------------------------------------------------------
// Kernel 5: reduce 16 T-chunk partials -> context. grid = B, block 256.
// ---------------------------------------------------------------------------
__global__ __launch_bounds__(256) void k_ctx_reduce(const float* __restrict__ partial,
                                                    float* __restrict__ out) {
  const int b = blockIdx.x;
  for (int h = threadIdx.x; h < H; h += 256) {
    float acc = 0.f;
#pragma unroll
    for (int tc = 0; tc < 16; ++tc) acc += partial[((size_t)b * 16 + tc) * H + h];
    out[(size_t)b * H + h] = acc;
  }
}

extern "C" void kernel_launch(void* const* d_in, const int* in_sizes, int n_in,
                              void* d_out, int out_size, void* d_ws, size_t ws_size,
                              hipStream_t stream) {
  const float*         enc  = (const float*)d_in[0];         // (B,T,H)
  const float*         dec  = (const float*)d_in[1];         // (B,H)
  const unsigned char* mask = (const unsigned char*)d_in[2]; // (B,T) bool
  const float*         Wa   = (const float*)d_in[3];         // (H,U)
  float* out = (float*)d_out;                                 // (B,H)

  float* ws      = (float*)d_ws;
  float* score   = ws;                       // B*T
  float* attn    = score + (size_t)B * T;    // B*T
  float* v       = attn + (size_t)B * T;     // B*H
  float* partial = v + (size_t)B * H;        // B*16*H

  k_project<<<dim3(H / 64), 256, 0, stream>>>(dec, Wa, v);
  k_score<<<dim3(T / 64, B), 256, 0, stream>>>(enc, v, mask, score);
  k_softmax<<<dim3(B), 256, 0, stream>>>(score, attn);
  k_ctx_partial<<<dim3(H / 256, T / 128, B), 256, 0, stream>>>(enc, attn, mask, partial);
  k_ctx_reduce<<<dim3(B), 256, 0, stream>>>(partial, out);
}